// DiffQuantumSimulator_2534030705108
// MI455X (gfx1250) — compile-verified
//
#include <hip/hip_runtime.h>
#include <math.h>

#define NQ     18
#define DIM    262144   // 2^18
#define BATCH  8
#define LAYERS 4

typedef float v2f __attribute__((ext_vector_type(2)));
typedef float v8f __attribute__((ext_vector_type(8)));

// ---------------- CDNA5 async global<->LDS helpers (ASYNCcnt path) -----------
__device__ __forceinline__ unsigned lds_off(const void* p) {
  // flat address of an LDS location keeps the LDS byte offset in the low 32 bits
  return (unsigned)(unsigned long long)p;
}
__device__ __forceinline__ void async_load_b128(unsigned loff, const void* g) {
  asm volatile("global_load_async_to_lds_b128 %0, %1, off"
               :: "v"(loff), "v"(g) : "memory");
}
__device__ __forceinline__ void async_store_b128(const void* g, unsigned loff) {
  asm volatile("global_store_async_from_lds_b128 %0, %1, off"
               :: "v"(g), "v"(loff) : "memory");
}
__device__ __forceinline__ void wait_async0() {
  asm volatile("s_wait_asynccnt 0" ::: "memory");
}

// -------- fused 4-qubit RX gate: U = (RX(beta))^{tensor 4}, 16x16 complex ----
// U[m][n] = c^(4-h) * (-i s)^h, h = popcount(m^n)  -> real if h even, imag if h odd.
struct GateFrags {
  v2f Ur[4];   // A-fragments of Re(U), k-th 16x4 column slice
  v2f Ui[4];   // A-fragments of Im(U)
  v2f UiN[4];  // A-fragments of -Im(U)  (f32 WMMA has no A-neg modifier)
};

__device__ __forceinline__ GateFrags make_gate_frags(float c, float s, int lane) {
  float t[5];
  t[0] = c*c*c*c; t[1] = c*c*c*s; t[2] = c*c*s*s; t[3] = c*s*s*s; t[4] = s*s*s*s;
  int m  = lane & 15;
  int kb = (lane < 16) ? 0 : 2;   // A layout: lanes 0-15 hold K=0,1 ; lanes 16-31 hold K=2,3
  GateFrags g;
#pragma unroll
  for (int k = 0; k < 4; ++k) {
    float vr[2], vi[2];
#pragma unroll
    for (int e = 0; e < 2; ++e) {
      int n = 4*k + kb + e;
      int h = __popc(m ^ n);
      float val = t[h];
      float re = 0.f, im = 0.f;
      if ((h & 1) == 0) re = (h & 2) ? -val : val;   // (-i)^h = +1 / -1
      else              im = (h & 2) ?  val : -val;  // (-i)^h = -i / +i
      vr[e] = re; vi[e] = im;
    }
    g.Ur[k]  = (v2f){ vr[0],  vr[1]};
    g.Ui[k]  = (v2f){ vi[0],  vi[1]};
    g.UiN[k] = (v2f){-vi[0], -vi[1]};
  }
  return g;
}

// Apply fused gate on LDS bit positions [j .. j+3] of a 4096-amplitude LDS tile.
// addr(m, col) = ((col>>j) << (j+4)) + (m << j) + (col & ((1<<j)-1)), 256 columns,
// 16 WMMA tiles per block, 2 per wave (8 waves). Tiles read/write disjoint sets.
__device__ __forceinline__ void apply_group_wmma(float2* lds, int j,
                                                 const GateFrags& g,
                                                 int lane, int wave) {
#pragma unroll
  for (int tt = 0; tt < 2; ++tt) {
    int tile = wave * 2 + tt;
    int col  = tile * 16 + (lane & 15);
    int kb   = (lane < 16) ? 0 : 2;
    int lom  = (1 << j) - 1;
    int base = ((col >> j) << (j + 4)) + (col & lom);
    v2f Br[4], Bi[4];
#pragma unroll
    for (int k = 0; k < 4; ++k) {  // B layout mirrors A: lanes 0-15 rows k,k+1; 16-31 rows k+2,k+3
      float2 e0 = lds[base + ((4*k + kb)     << j)];
      float2 e1 = lds[base + ((4*k + kb + 1) << j)];
      Br[k] = (v2f){e0.x, e1.x};
      Bi[k] = (v2f){e0.y, e1.y};
    }
    v8f Dr = (v8f){0,0,0,0,0,0,0,0};
    v8f Di = (v8f){0,0,0,0,0,0,0,0};
#pragma unroll
    for (int k = 0; k < 4; ++k) {
      Dr = __builtin_amdgcn_wmma_f32_16x16x4_f32(false, g.Ur[k],  false, Br[k], (short)0, Dr, false, false);
      Dr = __builtin_amdgcn_wmma_f32_16x16x4_f32(false, g.UiN[k], false, Bi[k], (short)0, Dr, false, false);
      Di = __builtin_amdgcn_wmma_f32_16x16x4_f32(false, g.Ur[k],  false, Bi[k], (short)0, Di, false, false);
      Di = __builtin_amdgcn_wmma_f32_16x16x4_f32(false, g.Ui[k],  false, Br[k], (short)0, Di, false, false);
    }
    int mb = (lane < 16) ? 0 : 8;  // D layout: lanes 0-15 -> M=vgpr, lanes 16-31 -> M=vgpr+8
#pragma unroll
    for (int r = 0; r < 8; ++r)
      lds[base + ((r + mb) << j)] = make_float2(Dr[r], Di[r]);
  }
}

// cut(d) = sum_u popcount(mask_u & (bit_u(d) ? ~d : d));  hp_diag == cut exactly.
__device__ __forceinline__ int cut_of(int d, const int* smask) {
  int c = 0;
#pragma unroll
  for (int u = 0; u < NQ; ++u) {
    int sel = ((d >> (NQ - 1 - u)) & 1) ? ~d : d;
    c += __popc(smask[u] & sel);
  }
  return c;
}

// -------- setup kernels --------------------------------------------------
__global__ void qaoa_masks(const float* __restrict__ adj, int* __restrict__ masks) {
  int b = blockIdx.x;
  int u = threadIdx.x;
  if (u < NQ) {
    int m = 0;
    for (int v = u + 1; v < NQ; ++v)
      if (adj[b*NQ*NQ + u*NQ + v] > 0.5f) m |= 1 << (NQ - 1 - v);
    masks[b*NQ + u] = m;
  }
}

__global__ __launch_bounds__(256) void qaoa_init(float4* __restrict__ state4) {
  int idx = blockIdx.x * 1024 + threadIdx.x;
  const float a = 0.001953125f;  // 1/sqrt(2^18) = 1/512
#pragma unroll
  for (int i = 0; i < 4; ++i)
    state4[idx + i * 256] = make_float4(a, 0.f, a, 0.f);
}

// -------- one layer = low pass (phase + qubit bits 0..11) ----------------
__global__ __launch_bounds__(256) void qaoa_layer_low(float2* __restrict__ state,
                                                      const int* __restrict__ masks,
                                                      const float* __restrict__ betas,
                                                      int layer) {
  __shared__ float2 lds[4096];
  __shared__ int smask[NQ];
  int b     = blockIdx.x >> 6;
  int chunk = blockIdx.x & 63;
  int tid   = threadIdx.x;
  if (tid < NQ) smask[tid] = masks[b*NQ + tid];
  float beta = betas[b*LAYERS + layer];
  float cg = cosf(beta), sg = sinf(beta);
  int lane = tid & 31, wave = tid >> 5;
  GateFrags g = make_gate_frags(cg, sg, lane);
  __syncthreads();
  int base = b * DIM + chunk * 4096;
#pragma unroll 2
  for (int i = 0; i < 8; ++i) {           // load 2 amplitudes + phase exp(-i*cut)
    int l = 2 * tid + i * 512;
    float4 v = *(const float4*)(const void*)&state[base + l];
    int d0 = chunk * 4096 + l;
    int c0 = cut_of(d0, smask);
    int c1 = cut_of(d0 + 1, smask);
    float s0, co0, s1, co1;
    sincosf((float)c0, &s0, &co0);
    sincosf((float)c1, &s1, &co1);
    float4 r;
    r.x = v.x * co0 + v.y * s0;  r.y = v.y * co0 - v.x * s0;
    r.z = v.z * co1 + v.w * s1;  r.w = v.w * co1 - v.z * s1;
    *(float4*)(void*)&lds[l] = r;
  }
  __syncthreads();
  apply_group_wmma(lds, 0, g, lane, wave); __syncthreads();
  apply_group_wmma(lds, 4, g, lane, wave); __syncthreads();
  apply_group_wmma(lds, 8, g, lane, wave); __syncthreads();
#pragma unroll
  for (int i = 0; i < 8; ++i) {           // LDS -> global, async DMA path
    int l = 2 * tid + i * 512;
    async_store_b128(&state[base + l], lds_off(&lds[l]));
  }
  wait_async0();
}

// -------- high pass: qubit bits 12..17 (gather 64 strided slabs) ---------
__global__ __launch_bounds__(256) void qaoa_layer_high(float2* __restrict__ state,
                                                       const float* __restrict__ betas,
                                                       int layer) {
  __shared__ float2 lds[4096];
  int b   = blockIdx.x >> 6;
  int grp = blockIdx.x & 63;
  int tid = threadIdx.x;
  float beta = betas[b*LAYERS + layer];
  float cg = cosf(beta), sg = sinf(beta);
  int lane = tid & 31, wave = tid >> 5;
  int base = b * DIM;
  // global d = hi*4096 + grp*64 + lo  ->  lds index l = hi*64 + lo
#pragma unroll
  for (int i = 0; i < 8; ++i) {           // global -> LDS, async DMA path
    int l = 2 * tid + i * 512;
    async_load_b128(lds_off(&lds[l]),
                    &state[base + (l >> 6) * 4096 + grp * 64 + (l & 63)]);
  }
  GateFrags g = make_gate_frags(cg, sg, lane);  // overlaps with async loads
  wait_async0();
  __syncthreads();
  apply_group_wmma(lds, 6, g, lane, wave);   // global bits 12..15 == LDS bits 6..9
  __syncthreads();
#pragma unroll
  for (int p = 10; p <= 11; ++p) {           // global bits 16,17 scalar butterflies
#pragma unroll 2
    for (int i = 0; i < 8; ++i) {
      int t  = tid + i * 256;                // 0..2047 pair ids
      int l0 = ((t >> p) << (p + 1)) | (t & ((1 << p) - 1));
      int l1 = l0 | (1 << p);
      float2 a0 = lds[l0], a1 = lds[l1];
      lds[l0] = make_float2(cg * a0.x + sg * a1.y,  cg * a0.y - sg * a1.x);
      lds[l1] = make_float2(sg * a0.y + cg * a1.x, -sg * a0.x + cg * a1.y);
    }
    __syncthreads();
  }
#pragma unroll
  for (int i = 0; i < 8; ++i) {           // LDS -> global, async DMA path
    int l = 2 * tid + i * 512;
    async_store_b128(&state[base + (l >> 6) * 4096 + grp * 64 + (l & 63)],
                     lds_off(&lds[l]));
  }
  wait_async0();
}

// -------- energy reduction ----------------------------------------------
__global__ void qaoa_zero(float* out) {
  if (threadIdx.x < BATCH) out[threadIdx.x] = 0.f;
}

__global__ __launch_bounds__(256) void qaoa_energy(const float2* __restrict__ state,
                                                   const int* __restrict__ masks,
                                                   float* __restrict__ out) {
  __shared__ int smask[NQ];
  __shared__ float red[256];
  int b   = blockIdx.x >> 4;
  int seg = blockIdx.x & 15;
  int tid = threadIdx.x;
  if (tid < NQ) smask[tid] = masks[b*NQ + tid];
  __syncthreads();
  int base = b * DIM + seg * 16384;
  float acc = 0.f;
  for (int i = 0; i < 32; ++i) {
    int l = 2 * tid + i * 512;
    float4 v = *(const float4*)(const void*)&state[base + l];
    int d0 = seg * 16384 + l;
    acc += (v.x * v.x + v.y * v.y) * (float)cut_of(d0, smask);
    acc += (v.z * v.z + v.w * v.w) * (float)cut_of(d0 + 1, smask);
  }
  red[tid] = acc;
  __syncthreads();
  for (int s = 128; s > 0; s >>= 1) {
    if (tid < s) red[tid] += red[tid + s];
    __syncthreads();
  }
  if (tid == 0) atomicAdd(&out[b], red[0]);
}

// -------- launch ---------------------------------------------------------
extern "C" void kernel_launch(void* const* d_in, const int* in_sizes, int n_in,
                              void* d_out, int out_size, void* d_ws, size_t ws_size,
                              hipStream_t stream) {
  const float* betas = (const float*)d_in[0];  // (8,4)
  const float* adj   = (const float*)d_in[1];  // (8,18,18)
  float* out = (float*)d_out;                  // (8,)
  float2* state = (float2*)d_ws;                                  // 16 MB
  int* masks = (int*)((char*)d_ws + (size_t)BATCH * DIM * sizeof(float2));

  qaoa_masks<<<BATCH, 32, 0, stream>>>(adj, masks);
  qaoa_init<<<1024, 256, 0, stream>>>((float4*)state);
  for (int t = 0; t < LAYERS; ++t) {
    qaoa_layer_low <<<BATCH * 64, 256, 0, stream>>>(state, masks, betas, t);
    qaoa_layer_high<<<BATCH * 64, 256, 0, stream>>>(state, betas, t);
  }
  qaoa_zero<<<1, 32, 0, stream>>>(out);
  qaoa_energy<<<BATCH * 16, 256, 0, stream>>>(state, masks, out);
}